// Residual_InteractionBlock_12249246728954
// MI455X (gfx1250) — compile-verified
//
#include <hip/hip_runtime.h>
#include <hip/hip_bf16.h>

// ---------------------------------------------------------------------------
// MACE-style interaction block for MI455X (gfx1250, wave32, WMMA).
// N=20000 nodes, E=320000 edges, C=64, L_DIMS=(1,3,5) -> 9, msg width 576.
//
// Key choices:
//  * Radial MLP has no nonlinearity -> fold W_r0..W_r3 (+ 1/sqrt scales) into
//    one 8x192 composite matrix once per launch (13.4 GFLOP -> 1 GFLOP).
//  * All GEMMs use V_WMMA_F32_16X16X4_F32 (one 16x16 tile per wave).
//  * Edge scatter-add: msg (46 MB) + h (5 MB) fit the 192 MB L2, so the
//    184M f32 atomics resolve in L2.
// ---------------------------------------------------------------------------

typedef float v2f __attribute__((ext_vector_type(2)));
typedef float v8f __attribute__((ext_vector_type(8)));

#define NN       20000
#define EE       320000
#define CC       64
#define NATTR    10
#define MSGW     576            // C * (1+3+5)
#define OUTSZ    (NN * MSGW)    // 11,520,000 floats (first tuple element)

__device__ __forceinline__ v8f wmma_f32(v2f a, v2f b, v8f c) {
  // 8 args: (neg_a, A, neg_b, B, c_mod, C, reuse_a, reuse_b)
  return __builtin_amdgcn_wmma_f32_16x16x4_f32(false, a, false, b, (short)0, c,
                                               false, false);
}

// ---------------------------------------------------------------------------
// Kernel 0: fold radial weights: Wcomp(8x192) = (W0/sqrt8)(W1/8)(W2/8)(W3/8)
// ---------------------------------------------------------------------------
__global__ void __launch_bounds__(256)
radial_fold_kernel(const float* __restrict__ W0, const float* __restrict__ W1,
                   const float* __restrict__ W2, const float* __restrict__ W3,
                   float* __restrict__ Wcomp) {
  __shared__ float t1[8 * 64];
  __shared__ float t2[8 * 64];
  const int tid = threadIdx.x;
  const float s1 = 0.04419417382f;          // 1/(sqrt(8)*8)
  for (int idx = tid; idx < 512; idx += 256) {
    int i = idx >> 6, j = idx & 63;
    float a = 0.f;
    for (int k = 0; k < 64; ++k) a += W0[i * 64 + k] * W1[k * 64 + j];
    t1[idx] = a * s1;
  }
  __syncthreads();
  for (int idx = tid; idx < 512; idx += 256) {
    int i = idx >> 6, j = idx & 63;
    float a = 0.f;
    for (int k = 0; k < 64; ++k) a += t1[i * 64 + k] * W2[k * 64 + j];
    t2[idx] = a * 0.125f;
  }
  __syncthreads();
  for (int idx = tid; idx < 8 * 192; idx += 256) {
    int i = idx / 192, j = idx % 192;
    float a = 0.f;
    for (int k = 0; k < 64; ++k) a += t2[i * 64 + k] * W3[k * 192 + j];
    Wcomp[idx] = a * 0.125f;
  }
}

// ---------------------------------------------------------------------------
// Kernel 1: h = node_feats (N x 64) @ (W_lin1 / 8).  WMMA f32 16x16x4.
// Block = 4 waves; wave w computes tile (blockIdx.x, w) of the N x 64 result.
// ---------------------------------------------------------------------------
__global__ void __launch_bounds__(128)
gemm_h_kernel(const float* __restrict__ A, const float* __restrict__ B,
              float* __restrict__ H) {
  const int wave = threadIdx.x >> 5;
  const int lane = threadIdx.x & 31;
  const int hi = lane >> 4;          // 0: K pair {0,1}; 1: K pair {2,3}
  const int l15 = lane & 15;
  const int aRow = blockIdx.x * 16 + l15;
  const int bCol = wave * 16 + l15;
  v8f acc = {};
  for (int k0 = 0; k0 < 64; k0 += 4) {
    const int ka = k0 + 2 * hi;
    v2f a, b;
    a.x = A[aRow * 64 + ka];
    a.y = A[aRow * 64 + ka + 1];
    b.x = B[ka * 64 + bCol];
    b.y = B[(ka + 1) * 64 + bCol];
    acc = wmma_f32(a, b, acc);
  }
#pragma unroll
  for (int r = 0; r < 8; ++r) {
    const int row = blockIdx.x * 16 + r + hi * 8;
    H[row * 64 + bCol] = acc[r] * 0.125f;   // / sqrt(64)
  }
}

// ---------------------------------------------------------------------------
// Kernel 2: zero msg accumulator
// ---------------------------------------------------------------------------
__global__ void zero_kernel(float* __restrict__ p, int n) {
  int i = blockIdx.x * blockDim.x + threadIdx.x;
  if (i < n) p[i] = 0.f;
}

// ---------------------------------------------------------------------------
// Kernel 3: edge tensor-product + scatter-add.
// Block = 8 waves; wave handles one edge. Stage h[sender] (64), w (192),
// y (9) into LDS, then 18 coalesced waves of 32 f32 atomics (576 cols/edge).
// mji column j: l=0 -> c (j<64); l=1 -> 64+3c+m; l=2 -> 256+5c+m.
// ---------------------------------------------------------------------------
__global__ void __launch_bounds__(256)
edge_kernel(const float* __restrict__ h, const float* __restrict__ edge_attrs,
            const float* __restrict__ edge_feats,
            const long long* __restrict__ eidx,
            const float* __restrict__ Wcomp, float* __restrict__ msg) {
  __shared__ float h_s[8][64];
  __shared__ float w_s[8][192];
  __shared__ float y_s[8][9];
  const int wave = threadIdx.x >> 5;
  const int lane = threadIdx.x & 31;
  const int e = blockIdx.x * 8 + wave;
  const long long snd = eidx[e];
  const long long rcv = eidx[EE + e];

  // stage h[sender]
  const float* hs = h + snd * 64;
  h_s[wave][lane]      = hs[lane];
  h_s[wave][lane + 32] = hs[lane + 32];

  // radial: w = edge_feats[e] (8) @ Wcomp (8x192), each lane makes 6 outputs
  float ef[8];
#pragma unroll
  for (int k = 0; k < 8; ++k) ef[k] = edge_feats[e * 8 + k];
#pragma unroll
  for (int t = 0; t < 6; ++t) {
    const int j = lane + t * 32;
    float a = 0.f;
#pragma unroll
    for (int k = 0; k < 8; ++k) a += ef[k] * Wcomp[k * 192 + j];
    w_s[wave][j] = a;
  }
  if (lane < 9) y_s[wave][lane] = edge_attrs[e * 9 + lane];
  __syncthreads();

  const int base = (int)rcv * MSGW;
#pragma unroll
  for (int it = 0; it < 18; ++it) {
    const int j = it * 32 + lane;
    int l, c, m, offl;
    if (j < 64)        { l = 0; c = j;            m = 0;         offl = 0; }
    else if (j < 256)  { int u = j - 64;  c = u / 3; m = u - 3 * c; l = 1; offl = 1; }
    else               { int u = j - 256; c = u / 5; m = u - 5 * c; l = 2; offl = 4; }
    const float val = h_s[wave][c] * w_s[wave][l * 64 + c] * y_s[wave][offl + m];
    atomicAdd(&msg[base + j], val);
  }
}

// ---------------------------------------------------------------------------
// Kernel 4: out[n,v,:] = concat_l (msg_l(n,:,m) @ W_msg[l]) / (8*16).
// Rows are (n,m) pairs within one l-block; A element k sits at stride d.
// Row tiles: l=0 -> 1250, l=1 -> 3750, l=2 -> 6250 (total 11250).
// ---------------------------------------------------------------------------
__global__ void __launch_bounds__(128)
msg_out_kernel(const float* __restrict__ msg, const float* __restrict__ Wmsg,
               float* __restrict__ out) {
  const int t = blockIdx.x;
  const int wave = threadIdx.x >> 5;
  const int lane = threadIdx.x & 31;
  const int hi = lane >> 4;
  const int l15 = lane & 15;

  int l, d, offc, offm, rbase;
  if (t < 1250)      { l = 0; d = 1; offc = 0;   offm = 0; rbase = t * 16; }
  else if (t < 5000) { l = 1; d = 3; offc = 64;  offm = 1; rbase = (t - 1250) * 16; }
  else               { l = 2; d = 5; offc = 256; offm = 4; rbase = (t - 5000) * 16; }

  const float* Wm = Wmsg + l * 4096;
  const int bCol = wave * 16 + l15;

  const int aRow = rbase + l15;
  const int n_a = aRow / d, m_a = aRow % d;
  const float* aptr = msg + n_a * MSGW + offc + m_a;   // element k at aptr[k*d]

  v8f acc = {};
  for (int k0 = 0; k0 < 64; k0 += 4) {
    const int ka = k0 + 2 * hi;
    v2f a, b;
    a.x = aptr[ka * d];
    a.y = aptr[(ka + 1) * d];
    b.x = Wm[ka * 64 + bCol];
    b.y = Wm[(ka + 1) * 64 + bCol];
    acc = wmma_f32(a, b, acc);
  }
  const float s = 1.0f / 128.0f;     // 1/sqrt(64) * 1/AVG_NEIGH
#pragma unroll
  for (int r = 0; r < 8; ++r) {
    const int row = rbase + r + hi * 8;
    const int n = row / d, m = row % d;
    out[n * MSGW + bCol * 9 + offm + m] = acc[r] * s;
  }
}

// ---------------------------------------------------------------------------
// Kernel 5: sc = (h outer attr) (N x 640) @ W_skip (640 x 64) / sqrt(640).
// A fragment built on the fly: A[n, 10u+v] = h[n,u] * attr[n,v].
// ---------------------------------------------------------------------------
__global__ void __launch_bounds__(128)
skip_kernel(const float* __restrict__ h, const float* __restrict__ attrs,
            const float* __restrict__ Wsk, float* __restrict__ sc) {
  const int wave = threadIdx.x >> 5;
  const int lane = threadIdx.x & 31;
  const int hi = lane >> 4;
  const int l15 = lane & 15;
  const int row = blockIdx.x * 16 + l15;
  const int bCol = wave * 16 + l15;
  const float* hrow = h + row * 64;
  const float* arow = attrs + row * NATTR;

  v8f acc = {};
  for (int k0 = 0; k0 < 640; k0 += 4) {
    const int ka = k0 + 2 * hi;
    v2f a, b;
    a.x = hrow[ka / 10] * arow[ka % 10];
    a.y = hrow[(ka + 1) / 10] * arow[(ka + 1) % 10];
    b.x = Wsk[ka * 64 + bCol];
    b.y = Wsk[(ka + 1) * 64 + bCol];
    acc = wmma_f32(a, b, acc);
  }
  const float s = 0.03952847075f;    // 1/sqrt(640)
#pragma unroll
  for (int r = 0; r < 8; ++r) {
    const int rout = blockIdx.x * 16 + r + hi * 8;
    sc[rout * 64 + bCol] = acc[r] * s;
  }
}

// ---------------------------------------------------------------------------
extern "C" void kernel_launch(void* const* d_in, const int* in_sizes, int n_in,
                              void* d_out, int out_size, void* d_ws,
                              size_t ws_size, hipStream_t stream) {
  const float* node_attrs     = (const float*)d_in[0];
  const float* node_feats     = (const float*)d_in[1];
  const float* edge_attrs     = (const float*)d_in[2];
  const float* edge_feats     = (const float*)d_in[3];
  const long long* edge_index = (const long long*)d_in[4];   // int64 per ref
  const float* W_lin1         = (const float*)d_in[5];
  const float* W_r0           = (const float*)d_in[6];
  const float* W_r1           = (const float*)d_in[7];
  const float* W_r2           = (const float*)d_in[8];
  const float* W_r3           = (const float*)d_in[9];
  const float* W_msg          = (const float*)d_in[10];
  const float* W_skip         = (const float*)d_in[11];

  float* out = (float*)d_out;          // (N, 64, 9) flat
  float* sc  = out + OUTSZ;            // (N, 64) flat

  float* ws    = (float*)d_ws;
  float* Wcomp = ws;                   // 1536 floats
  float* h     = ws + 4096;            // N*64 = 1,280,000 floats
  float* msg   = h + NN * CC;          // N*576 = 11,520,000 floats

  radial_fold_kernel<<<1, 256, 0, stream>>>(W_r0, W_r1, W_r2, W_r3, Wcomp);
  gemm_h_kernel<<<NN / 16, 128, 0, stream>>>(node_feats, W_lin1, h);
  zero_kernel<<<(NN * MSGW + 255) / 256, 256, 0, stream>>>(msg, NN * MSGW);
  edge_kernel<<<EE / 8, 256, 0, stream>>>(h, edge_attrs, edge_feats,
                                          edge_index, Wcomp, msg);
  msg_out_kernel<<<11250, 128, 0, stream>>>(msg, W_msg, out);
  skip_kernel<<<NN / 16, 128, 0, stream>>>(h, node_attrs, W_skip, sc);
}